// GCN_57501022159512
// MI455X (gfx1250) — compile-verified
//
#include <hip/hip_runtime.h>
#include <hip/hip_bf16.h>

typedef float v2f __attribute__((ext_vector_type(2)));
typedef float v8f __attribute__((ext_vector_type(8)));

#define BN_EPS 1e-5f

// ---------------------------------------------------------------------------
// Zero-fill
// ---------------------------------------------------------------------------
__global__ void zero_f32(float* __restrict__ p, long n) {
    long i = (long)blockIdx.x * blockDim.x + threadIdx.x;
    if (i < n) p[i] = 0.0f;
}

// ---------------------------------------------------------------------------
// Degree count: deg[col[e]] += 1  (deg buffer pre-zeroed)
// ---------------------------------------------------------------------------
__global__ void deg_count(const int* __restrict__ col, float* __restrict__ deg, int E) {
    int e = blockIdx.x * blockDim.x + threadIdx.x;
    if (e < E) atomicAdd(&deg[col[e]], 1.0f);
}

// dinv[i] = rsqrt(deg[i] + 1)   (+1 = self loop)
__global__ void make_dinv(float* __restrict__ deg, int N) {
    int i = blockIdx.x * blockDim.x + threadIdx.x;
    if (i < N) deg[i] = rsqrtf(deg[i] + 1.0f);
}

// ---------------------------------------------------------------------------
// GEMM via V_WMMA_F32_16X16X4_F32. One wave = one 16x16 output tile.
// A: [N, Fin] row-major, W: [Fin, Fout] row-major, out: [N, Fout].
// N must be a multiple of 16 (100000 = 6250*16). EXEC is uniform per wave.
// ---------------------------------------------------------------------------
__global__ void gemm_wmma_f32(const float* __restrict__ A,
                              const float* __restrict__ W,
                              float* __restrict__ out,
                              int N, int Fin, int Fout, int totalTiles) {
    const int wavesPerBlock = blockDim.x >> 5;
    const int waveId = threadIdx.x >> 5;
    const int lane   = threadIdx.x & 31;
    const int tile   = blockIdx.x * wavesPerBlock + waveId;
    if (tile >= totalTiles) return;              // wave-uniform exit

    const int tilesN = Fout >> 4;
    const int mt = tile / tilesN;
    const int nt = tile - mt * tilesN;
    const int row0 = mt << 4;
    const int col0 = nt << 4;
    const int lmod = lane & 15;
    const bool hi  = lane >= 16;

    v8f acc = {};
    for (int k = 0; k < Fin; k += 4) {
        // A-matrix 16x4 f32 layout: lanes0-15: K=k,k+1 (v0,v1); lanes16-31: K=k+2,k+3
        const int ka = k + (hi ? 2 : 0);
        v2f a, b;
        const float* arow = A + (long)(row0 + lmod) * Fin + ka;
        a.x = arow[0];
        a.y = arow[1];
        // B-matrix 4x16 f32: lanes0-15 carry rows k,k+1; lanes16-31 rows k+2,k+3
        b.x = W[(long)ka * Fout + col0 + lmod];
        b.y = W[(long)(ka + 1) * Fout + col0 + lmod];
        acc = __builtin_amdgcn_wmma_f32_16x16x4_f32(
            /*neg_a=*/false, a, /*neg_b=*/false, b,
            /*c_mod=*/(short)0, acc, /*reuse_a=*/false, /*reuse_b=*/false);
    }

    // C/D layout: VGPR r -> M = r (lanes 0-15) or M = r+8 (lanes 16-31), N = lane%16
    const int rbase = row0 + (hi ? 8 : 0);
#pragma unroll
    for (int r = 0; r < 8; ++r) {
        out[(long)(rbase + r) * Fout + col0 + lmod] = acc[r];
    }
}

// ---------------------------------------------------------------------------
// Edge scatter: agg[col[e]][:] += h[row[e]][:] * dinv[row]*dinv[col]
// One wave per edge; F=128 -> 32 lanes x float4 = exactly one row.
// ---------------------------------------------------------------------------
__global__ void edge_scatter(const float* __restrict__ h,
                             const float* __restrict__ dinv,
                             const int* __restrict__ row,
                             const int* __restrict__ col,
                             float* __restrict__ agg,
                             int E, int F) {
    const int e = blockIdx.x * (blockDim.x >> 5) + (threadIdx.x >> 5);
    if (e >= E) return;
    const int lane = threadIdx.x & 31;
    const int r = row[e];
    const int c = col[e];
    const float norm = dinv[r] * dinv[c];

    const float4 v = ((const float4*)(h + (long)r * F))[lane];
    float* dst = agg + (long)c * F + lane * 4;
    atomicAdd(dst + 0, v.x * norm);
    atomicAdd(dst + 1, v.y * norm);
    atomicAdd(dst + 2, v.z * norm);
    atomicAdd(dst + 3, v.w * norm);
}

// ---------------------------------------------------------------------------
// Epilogue: agg[i,f] = relu(agg[i,f] + h[i,f]*dinv[i]^2 + b[f])   (in place)
// ---------------------------------------------------------------------------
__global__ void finalize_relu(float* __restrict__ agg,
                              const float* __restrict__ h,
                              const float* __restrict__ dinv,
                              const float* __restrict__ b,
                              int N, int F) {
    long i = (long)blockIdx.x * blockDim.x + threadIdx.x;
    if (i >= (long)N * F) return;
    const int node = (int)(i / F);
    const int f = (int)(i - (long)node * F);
    const float di = dinv[node];
    const float v = agg[i] + h[i] * (di * di) + b[f];
    agg[i] = fmaxf(v, 0.0f);
}

// ---------------------------------------------------------------------------
// BN stats: per-feature sum and sum-of-squares. blockDim.x == F (128);
// each block register-accumulates a node stripe, then one atomic per feature.
// ---------------------------------------------------------------------------
__global__ void bn_stats(const float* __restrict__ x, float* __restrict__ sums,
                         int N, int F) {
    const int f = threadIdx.x;
    float s = 0.0f, s2 = 0.0f;
    for (int n = blockIdx.x; n < N; n += gridDim.x) {
        const float v = x[(long)n * F + f];
        s += v;
        s2 += v * v;
    }
    atomicAdd(&sums[f], s);
    atomicAdd(&sums[F + f], s2);
}

// y = gamma*(x-mean)*rsqrt(var+eps) + beta
__global__ void bn_apply(const float* __restrict__ x, float* __restrict__ y,
                         const float* __restrict__ sums,
                         const float* __restrict__ gamma,
                         const float* __restrict__ beta,
                         int N, int F) {
    long i = (long)blockIdx.x * blockDim.x + threadIdx.x;
    if (i >= (long)N * F) return;
    const int f = (int)(i % F);
    const float invN = 1.0f / (float)N;
    const float mean = sums[f] * invN;
    const float var  = sums[F + f] * invN - mean * mean;
    y[i] = gamma[f] * (x[i] - mean) * rsqrtf(var + BN_EPS) + beta[f];
}

// ---------------------------------------------------------------------------
// Launch
// ---------------------------------------------------------------------------
extern "C" void kernel_launch(void* const* d_in, const int* in_sizes, int n_in,
                              void* d_out, int out_size, void* d_ws, size_t ws_size,
                              hipStream_t stream) {
    const float* x     = (const float*)d_in[0];
    const int*   edges = (const int*)d_in[1];   // [2, E] int32
    const float* W1    = (const float*)d_in[2];
    const float* b1    = (const float*)d_in[3];
    const float* gamma = (const float*)d_in[4];
    const float* beta  = (const float*)d_in[5];
    const float* W2    = (const float*)d_in[6];
    const float* b2    = (const float*)d_in[7];
    float* out = (float*)d_out;

    const int F1 = 64;                   // input feature dim
    const int H  = in_sizes[3];          // 128 (hidden = len(b1))
    const int F2 = in_sizes[7];          // 128 (output = len(b2))
    const int N  = in_sizes[0] / F1;     // 100000
    const int E  = in_sizes[1] / 2;      // 1600000
    const int* row = edges;
    const int* col = edges + E;

    // workspace layout (floats)
    float* ws = (float*)d_ws;
    const long NP = ((long)N + 255) / 256 * 256;
    float* dinv  = ws;                    // N
    float* bufA  = ws + NP;               // N*H
    float* bufB  = bufA + (long)N * H;    // N*H
    float* stats = bufB + (long)N * H;    // 2*H

    const int T = 256;
    const long NH = (long)N * H;
    const long NF2 = (long)N * F2;

    // 1) degrees -> dinv
    zero_f32<<<(int)((N + T - 1) / T), T, 0, stream>>>(dinv, N);
    zero_f32<<<1, T, 0, stream>>>(stats, 2 * H);
    deg_count<<<(E + T - 1) / T, T, 0, stream>>>(col, dinv, E);
    make_dinv<<<(N + T - 1) / T, T, 0, stream>>>(dinv, N);

    // 2) h1 = x @ W1  (WMMA f32)
    {
        const int tiles = (N / 16) * (H / 16);
        gemm_wmma_f32<<<(tiles + 7) / 8, 256, 0, stream>>>(x, W1, bufA, N, F1, H, tiles);
    }

    // 3) agg1 = scatter(h1)
    zero_f32<<<(int)((NH + T - 1) / T), T, 0, stream>>>(bufB, NH);
    edge_scatter<<<(E + 7) / 8, 256, 0, stream>>>(bufA, dinv, row, col, bufB, E, H);

    // 4) relu(conv1) in bufB
    finalize_relu<<<(int)((NH + T - 1) / T), T, 0, stream>>>(bufB, bufA, dinv, b1, N, H);

    // 5) batch norm -> bufA
    bn_stats<<<512, H, 0, stream>>>(bufB, stats, N, H);
    bn_apply<<<(int)((NH + T - 1) / T), T, 0, stream>>>(bufB, bufA, stats, gamma, beta, N, H);

    // 6) h2 = hbn @ W2 (WMMA f32)
    {
        const int tiles = (N / 16) * (F2 / 16);
        gemm_wmma_f32<<<(tiles + 7) / 8, 256, 0, stream>>>(bufA, W2, bufB, N, H, F2, tiles);
    }

    // 7) agg2 -> d_out, then fused self-loop + bias + relu in place
    zero_f32<<<(int)((NF2 + T - 1) / T), T, 0, stream>>>(out, NF2);
    edge_scatter<<<(E + 7) / 8, 256, 0, stream>>>(bufB, dinv, row, col, out, E, F2);
    finalize_relu<<<(int)((NF2 + T - 1) / T), T, 0, stream>>>(out, bufB, dinv, b2, N, F2);
}